// GroupEmbedding_72980084294377
// MI455X (gfx1250) — compile-verified
//
#include <hip/hip_runtime.h>
#include <hip/hip_bf16.h>
#include <stdint.h>

#define EMB             128
#define WAVES_PER_BLOCK 8
#define BLOCK_THREADS   (WAVES_PER_BLOCK * 32)
#define SLOTS           8     // LDS ring slots per wave (power of 2)
#define PF_DEPTH        6     // async loads in flight per wave (< SLOTS-1)

// ---------------------------------------------------------------------------
// CDNA5 async global->LDS copy: one instruction per wave moves 32 lanes x 16B
// = 512B (one full embedding row). Tracked by ASYNCcnt, completes in order.
// ---------------------------------------------------------------------------
__device__ __forceinline__ void async_row_b128(uint32_t lds_addr, const float* gaddr) {
    asm volatile("global_load_async_to_lds_b128 %0, %1, off"
                 :: "v"(lds_addr), "v"(gaddr)
                 : "memory");
}

__device__ __forceinline__ void wait_asynccnt_le(int outstanding) {
    switch (outstanding) {
        case 0:  asm volatile("s_wait_asynccnt 0x0" ::: "memory"); break;
        case 1:  asm volatile("s_wait_asynccnt 0x1" ::: "memory"); break;
        case 2:  asm volatile("s_wait_asynccnt 0x2" ::: "memory"); break;
        case 3:  asm volatile("s_wait_asynccnt 0x3" ::: "memory"); break;
        case 4:  asm volatile("s_wait_asynccnt 0x4" ::: "memory"); break;
        case 5:  asm volatile("s_wait_asynccnt 0x5" ::: "memory"); break;
        default: asm volatile("s_wait_asynccnt 0x6" ::: "memory"); break;
    }
}

__device__ __forceinline__ void atomic_add_f32(float* p, float v) {
    __hip_atomic_fetch_add(p, v, __ATOMIC_RELAXED, __HIP_MEMORY_SCOPE_AGENT);
}

__global__ void ge_zero_out(float* __restrict__ out, int n) {
    int i = blockIdx.x * blockDim.x + threadIdx.x;
    if (i < n) out[i] = 0.0f;
}

__global__ __launch_bounds__(BLOCK_THREADS)
void ge_main(const int* __restrict__ user_ids,
             const int* __restrict__ user_group_ids,
             const int* __restrict__ behavior_item_ids,
             const float* __restrict__ behavior_counts,
             const int* __restrict__ behavior_user_ids,
             const float* __restrict__ user_table,
             const float* __restrict__ item_table,
             float* __restrict__ out,
             int n_users, int n_beh) {
    __align__(16) __shared__ float smem[WAVES_PER_BLOCK][SLOTS][EMB];

    const int wib  = threadIdx.x >> 5;
    const int lane = threadIdx.x & 31;
    const int u    = blockIdx.x * WAVES_PER_BLOCK + wib;
    if (u >= n_users) return;

    // behavior_user_ids is sorted: binary search this user's contiguous range.
    int lo = 0, hi = n_beh;
    while (lo < hi) { int mid = (lo + hi) >> 1; if (behavior_user_ids[mid] <  u) lo = mid + 1; else hi = mid; }
    const int start = lo;
    hi = n_beh;
    while (lo < hi) { int mid = (lo + hi) >> 1; if (behavior_user_ids[mid] <= u) lo = mid + 1; else hi = mid; }
    const int end = lo;

    float* const slots = &smem[wib][0][0];
    // low 32 bits of a generic pointer to LDS == workgroup-relative LDS byte address
    const uint32_t lds_lane_base = (uint32_t)(size_t)slots + (uint32_t)(lane * 16);

    float a0 = 0.f, a1 = 0.f, a2 = 0.f, a3 = 0.f;

    // Pipeline prologue: kick off up to PF_DEPTH async row fetches.
    #pragma unroll
    for (int k = 0; k < PF_DEPTH; ++k) {
        int j = start + k;
        if (j < end) {
            int bid = behavior_item_ids[j];
            async_row_b128(lds_lane_base + (uint32_t)((k & (SLOTS - 1)) * EMB * 4),
                           item_table + (size_t)bid * EMB + lane * 4);
        }
    }

    for (int j = start; j < end; ++j) {
        const int idx = j - start;
        const int jn  = j + PF_DEPTH;
        if (jn < end) {
            int bid = behavior_item_ids[jn];
            async_row_b128(lds_lane_base + (uint32_t)(((idx + PF_DEPTH) & (SLOTS - 1)) * EMB * 4),
                           item_table + (size_t)bid * EMB + lane * 4);
        }
        int outstanding = end - 1 - j;          // async ops for indices > j
        if (outstanding > PF_DEPTH) outstanding = PF_DEPTH;
        wait_asynccnt_le(outstanding);          // row j now resident in LDS

        const float4 v = *(const float4*)(slots + (idx & (SLOTS - 1)) * EMB + lane * 4);
        const float  c = behavior_counts[j];
        a0 = fmaf(c, v.x, a0);
        a1 = fmaf(c, v.y, a1);
        a2 = fmaf(c, v.z, a2);
        a3 = fmaf(c, v.w, a3);
    }

    // user embedding with padding_idx = 0
    const int uid = user_ids[u];
    float u0 = 0.f, u1 = 0.f, u2 = 0.f, u3 = 0.f;
    if (uid != 0) {
        const float4 uv = *(const float4*)(user_table + (size_t)uid * EMB + lane * 4);
        u0 = uv.x; u1 = uv.y; u2 = uv.z; u3 = uv.w;
    }

    const int g = user_group_ids[u];
    float* op = out + (size_t)g * EMB + lane * 4;
    atomic_add_f32(op + 0, a0 * u0);
    atomic_add_f32(op + 1, a1 * u1);
    atomic_add_f32(op + 2, a2 * u2);
    atomic_add_f32(op + 3, a3 * u3);
}

extern "C" void kernel_launch(void* const* d_in, const int* in_sizes, int n_in,
                              void* d_out, int out_size, void* d_ws, size_t ws_size,
                              hipStream_t stream) {
    const int*   user_ids          = (const int*)  d_in[0];
    const int*   user_group_ids    = (const int*)  d_in[1];
    const int*   behavior_item_ids = (const int*)  d_in[2];
    const float* behavior_counts   = (const float*)d_in[3];
    const int*   behavior_user_ids = (const int*)  d_in[4];
    const float* user_table        = (const float*)d_in[5];
    const float* item_table        = (const float*)d_in[6];
    float*       out               = (float*)d_out;

    const int n_users = in_sizes[0];
    const int n_beh   = in_sizes[2];

    // zero the accumulation target (atomic adds follow)
    {
        int threads = 256;
        int blocks  = (out_size + threads - 1) / threads;
        hipLaunchKernelGGL(ge_zero_out, dim3(blocks), dim3(threads), 0, stream, out, out_size);
    }

    int blocks = (n_users + WAVES_PER_BLOCK - 1) / WAVES_PER_BLOCK;
    hipLaunchKernelGGL(ge_main, dim3(blocks), dim3(BLOCK_THREADS), 0, stream,
                       user_ids, user_group_ids, behavior_item_ids, behavior_counts,
                       behavior_user_ids, user_table, item_table, out, n_users, n_beh);
}